// AttnDecoderRNN_5385888989778
// MI455X (gfx1250) — compile-verified
//
#include <hip/hip_runtime.h>
#include <hip/hip_bf16.h>
#include <math.h>

// ---------------------------------------------------------------------------
// AttnDecoderRNN single-step, H=1024, V=50257, L=40.  All f32.
// Bandwidth-bound (~240 MB weights/call -> ~10us @ 23.3 TB/s).
// Large GEMVs run on the f32 WMMA pipe (V_WMMA_F32_16X16X4_F32), 16 rows/wave,
// dual accumulators to avoid back-to-back WMMA RAW stalls, NT loads on the
// streaming weight matrix.
// ---------------------------------------------------------------------------

#define H_DIM 1024
#define V_DIM 50257
#define L_DIM 40
#define LSE_BLOCKS 128

typedef float v2f __attribute__((ext_vector_type(2)));
typedef float v4f __attribute__((ext_vector_type(4)));
typedef float v8f __attribute__((ext_vector_type(8)));

// ---------------------------------------------------------------------------
// y[0..M) = act( W(MxK) @ x(K) + bias ),  K % 16 == 0, rows 16B aligned.
// One wave per 16-row tile.  B operand = x broadcast across all 16 N columns,
// so every D column holds the identical y tile; lanes m==0 of each half write.
// A slot map (ISA 7.12.2): lanes0-15 vgpr{0,1}=K{0,1}; lanes16-31 = K{2,3}.
// B mirrors it. Dot is K-permutation invariant, so each lane streams a
// contiguous float4 of its row (lanes m and m+16 cover a contiguous 32B span).
// Two independent accumulators alternate so consecutive WMMAs have no RAW.
// ---------------------------------------------------------------------------
__global__ __launch_bounds__(256)
void gemv16_wmma_f32(const float* __restrict__ W, const float* __restrict__ x,
                     const float* __restrict__ bias, float* __restrict__ y,
                     int M, int K, int act /*0=none, 1=relu*/)
{
    const int lane = threadIdx.x & 31;
    const int wv   = threadIdx.x >> 5;
    const int tile = blockIdx.x * 8 + wv;
    const int numTiles = (M + 15) >> 4;
    if (tile >= numTiles) return;          // uniform per wave -> EXEC stays full

    const int row0 = tile << 4;
    const int hi   = lane >> 4;            // 0: K0/K1 slots, 1: K2/K3 slots
    const int m    = lane & 15;
    int arow = row0 + m;
    if (arow > M - 1) arow = M - 1;        // clamp loads; phantom rows not stored

    const float* wrow = W + (size_t)arow * (size_t)K + 4 * hi;
    const float* xptr = x + 4 * hi;

    v8f c0 = {0.f, 0.f, 0.f, 0.f, 0.f, 0.f, 0.f, 0.f};
    v8f c1 = {0.f, 0.f, 0.f, 0.f, 0.f, 0.f, 0.f, 0.f};

    for (int k = 0; k < K; k += 16) {
        // streaming weights: non-temporal (single-use, don't thrash L2/WGP$)
        v4f w0 = __builtin_nontemporal_load(
                     reinterpret_cast<const v4f*>(wrow + k));
        v4f w1 = __builtin_nontemporal_load(
                     reinterpret_cast<const v4f*>(wrow + k + 8));
        v4f x0 = *reinterpret_cast<const v4f*>(xptr + k);
        v4f x1 = *reinterpret_cast<const v4f*>(xptr + k + 8);

        v2f a00 = {w0.x, w0.y}, b00 = {x0.x, x0.y};
        v2f a01 = {w0.z, w0.w}, b01 = {x0.z, x0.w};
        v2f a10 = {w1.x, w1.y}, b10 = {x1.x, x1.y};
        v2f a11 = {w1.z, w1.w}, b11 = {x1.z, x1.w};

        c0 = __builtin_amdgcn_wmma_f32_16x16x4_f32(false, a00, false, b00,
                                                   (short)0, c0, false, false);
        c1 = __builtin_amdgcn_wmma_f32_16x16x4_f32(false, a01, false, b01,
                                                   (short)0, c1, false, false);
        c0 = __builtin_amdgcn_wmma_f32_16x16x4_f32(false, a10, false, b10,
                                                   (short)0, c0, false, false);
        c1 = __builtin_amdgcn_wmma_f32_16x16x4_f32(false, a11, false, b11,
                                                   (short)0, c1, false, false);
    }

    // D layout: VGPR r -> row r (lanes 0-15) / row r+8 (lanes 16-31),
    // all N columns identical.  Lane m==0 of each half writes 8 rows.
    if (m == 0) {
#pragma unroll
        for (int r = 0; r < 8; ++r) {
            int orow = row0 + r + 8 * hi;
            if (orow < M) {
                float v = c0[r] + c1[r] + bias[orow];
                if (act == 1) v = fmaxf(v, 0.f);
                y[orow] = v;
            }
        }
    }
}

// ---------------------------------------------------------------------------
// Prologue (1 block, 256 thr): x = emb[tok]; scores = W_attn @ [x;h] + b;
// softmax; attn_applied = aw @ enc; concat2 = [x ; attn_applied] into ws.
// ---------------------------------------------------------------------------
__global__ __launch_bounds__(256)
void attn_prologue(const int* __restrict__ input, const float* __restrict__ hidden,
                   const float* __restrict__ enc, const float* __restrict__ emb,
                   const float* __restrict__ Wa, const float* __restrict__ ba,
                   float* __restrict__ attn_out, float* __restrict__ concat2)
{
    __shared__ float s_sc[L_DIM];
    __shared__ float s_inv;
    const int tid  = threadIdx.x;
    const int lane = tid & 31;
    const int wvid = tid >> 5;

    const int tok = input[0];
    const float* x = emb + (size_t)tok * H_DIM;

    for (int i = tid; i < H_DIM; i += 256) concat2[i] = x[i];

    // 40 dot products of length 2048; one wave per score row, round-robin.
    for (int l = wvid; l < L_DIM; l += 8) {
        const float* wr = Wa + (size_t)l * (2 * H_DIM);
        float acc = 0.f;
        for (int k = lane; k < 2 * H_DIM; k += 32) {
            float xv = (k < H_DIM) ? x[k] : hidden[k - H_DIM];
            acc += wr[k] * xv;
        }
        for (int off = 16; off > 0; off >>= 1) acc += __shfl_xor(acc, off, 32);
        if (lane == 0) s_sc[l] = acc + ba[l];
    }
    __syncthreads();

    if (tid == 0) {
        float mx = s_sc[0];
        for (int l = 1; l < L_DIM; ++l) mx = fmaxf(mx, s_sc[l]);
        float sum = 0.f;
        for (int l = 0; l < L_DIM; ++l) { float e = expf(s_sc[l] - mx); s_sc[l] = e; sum += e; }
        s_inv = 1.0f / sum;
    }
    __syncthreads();
    if (tid < L_DIM) {
        float w = s_sc[tid] * s_inv;
        s_sc[tid]     = w;
        attn_out[tid] = w;
    }
    __syncthreads();

    for (int j = tid; j < H_DIM; j += 256) {
        float acc = 0.f;
#pragma unroll 8
        for (int l = 0; l < L_DIM; ++l) acc += s_sc[l] * enc[l * H_DIM + j];
        concat2[H_DIM + j] = acc;
    }
}

// ---------------------------------------------------------------------------
// GRU gating (elementwise over H).
// ---------------------------------------------------------------------------
__global__ __launch_bounds__(256)
void gru_gate(const float* __restrict__ gi, const float* __restrict__ gh,
              const float* __restrict__ h, float* __restrict__ hnew_out,
              float* __restrict__ hnew_ws)
{
    int i = blockIdx.x * blockDim.x + threadIdx.x;
    if (i >= H_DIM) return;
    float r = 1.f / (1.f + expf(-(gi[i] + gh[i])));
    float z = 1.f / (1.f + expf(-(gi[H_DIM + i] + gh[H_DIM + i])));
    float n = tanhf(gi[2 * H_DIM + i] + r * gh[2 * H_DIM + i]);
    float hv = h[i];
    float o = (1.f - z) * n + z * hv;
    hnew_out[i] = o;
    hnew_ws[i]  = o;
}

// ---------------------------------------------------------------------------
// Deterministic online logsumexp, 2 stages + in-place normalize.
// ---------------------------------------------------------------------------
__device__ __forceinline__ void lse_comb(float& m1, float& s1, float m2, float s2)
{
    if (s2 == 0.f) return;
    if (s1 == 0.f) { m1 = m2; s1 = s2; return; }
    float nm = fmaxf(m1, m2);
    s1 = s1 * expf(m1 - nm) + s2 * expf(m2 - nm);
    m1 = nm;
}

__global__ __launch_bounds__(256)
void lse_partial(const float* __restrict__ logits, int n, float* __restrict__ part)
{
    __shared__ float sm[256], ss[256];
    int tid = threadIdx.x;
    float m = -INFINITY, s = 0.f;
    for (int v = blockIdx.x * blockDim.x + tid; v < n; v += gridDim.x * blockDim.x) {
        float xv = logits[v];
        if (s == 0.f) { m = xv; s = 1.f; }
        else {
            float nm = fmaxf(m, xv);
            s = s * expf(m - nm) + expf(xv - nm);
            m = nm;
        }
    }
    sm[tid] = m; ss[tid] = s;
    __syncthreads();
    for (int off = 128; off > 0; off >>= 1) {
        if (tid < off) {
            float m1 = sm[tid], s1 = ss[tid];
            lse_comb(m1, s1, sm[tid + off], ss[tid + off]);
            sm[tid] = m1; ss[tid] = s1;
        }
        __syncthreads();
    }
    if (tid == 0) { part[2 * blockIdx.x] = sm[0]; part[2 * blockIdx.x + 1] = ss[0]; }
}

__global__ __launch_bounds__(128)
void lse_final(const float* __restrict__ part, int n, float* __restrict__ fin)
{
    __shared__ float sm[128], ss[128];
    int tid = threadIdx.x;
    float m = -INFINITY, s = 0.f;
    if (tid < n) { m = part[2 * tid]; s = part[2 * tid + 1]; }
    sm[tid] = m; ss[tid] = s;
    __syncthreads();
    for (int off = 64; off > 0; off >>= 1) {
        if (tid < off) {
            float m1 = sm[tid], s1 = ss[tid];
            lse_comb(m1, s1, sm[tid + off], ss[tid + off]);
            sm[tid] = m1; ss[tid] = s1;
        }
        __syncthreads();
    }
    if (tid == 0) { fin[0] = sm[0]; fin[1] = logf(ss[0]); }
}

__global__ __launch_bounds__(256)
void lse_normalize(float* __restrict__ out, const float* __restrict__ fin, int n)
{
    float M = fin[0], LS = fin[1];
    int stride = gridDim.x * blockDim.x;
    for (int v = blockIdx.x * blockDim.x + threadIdx.x; v < n; v += stride)
        out[v] = out[v] - M - LS;
}

// ---------------------------------------------------------------------------
// Launch.  Inputs (setup_inputs order):
//  0 input(i32,1) 1 hidden(H) 2 enc(L*H) 3 emb(V*H) 4 W_attn(L*2H) 5 b_attn(L)
//  6 W_comb(H*2H) 7 b_comb(H) 8 W_ih(3H*H) 9 b_ih(3H) 10 W_hh(3H*H) 11 b_hh(3H)
//  12 W_out(V*H) 13 b_out(V)
// d_out = [logits V | h_new H | attn_w L] (f32).
// ---------------------------------------------------------------------------
extern "C" void kernel_launch(void* const* d_in, const int* in_sizes, int n_in,
                              void* d_out, int out_size, void* d_ws, size_t ws_size,
                              hipStream_t stream)
{
    const int*   input  = (const int*)  d_in[0];
    const float* hidden = (const float*)d_in[1];
    const float* enc    = (const float*)d_in[2];
    const float* emb    = (const float*)d_in[3];
    const float* W_attn = (const float*)d_in[4];
    const float* b_attn = (const float*)d_in[5];
    const float* W_comb = (const float*)d_in[6];
    const float* b_comb = (const float*)d_in[7];
    const float* W_ih   = (const float*)d_in[8];
    const float* b_ih   = (const float*)d_in[9];
    const float* W_hh   = (const float*)d_in[10];
    const float* b_hh   = (const float*)d_in[11];
    const float* W_out  = (const float*)d_in[12];
    const float* b_out  = (const float*)d_in[13];

    float* out     = (float*)d_out;
    float* out_h   = out + V_DIM;
    float* out_aw  = out + V_DIM + H_DIM;

    float* ws      = (float*)d_ws;
    float* concat2 = ws;             // 2048
    float* g       = ws + 2048;      // 1024
    float* gi      = ws + 3072;      // 3072
    float* gh      = ws + 6144;      // 3072
    float* hn      = ws + 9216;      // 1024
    float* part    = ws + 10240;     // 2*LSE_BLOCKS
    float* fin     = ws + 10240 + 2 * LSE_BLOCKS;  // 2

    // 1) embed + attention + concat
    attn_prologue<<<1, 256, 0, stream>>>(input, hidden, enc, emb, W_attn, b_attn,
                                         out_aw, concat2);

    // 2) g = relu(W_comb @ concat2 + b_comb)   (1024 x 2048)
    {
        int tiles = (H_DIM + 15) / 16;
        gemv16_wmma_f32<<<(tiles + 7) / 8, 256, 0, stream>>>(
            W_comb, concat2, b_comb, g, H_DIM, 2 * H_DIM, 1);
    }
    // 3) gi = W_ih @ g + b_ih    (3072 x 1024)
    {
        int tiles = (3 * H_DIM + 15) / 16;
        gemv16_wmma_f32<<<(tiles + 7) / 8, 256, 0, stream>>>(
            W_ih, g, b_ih, gi, 3 * H_DIM, H_DIM, 0);
    }
    // 4) gh = W_hh @ h + b_hh    (3072 x 1024)
    {
        int tiles = (3 * H_DIM + 15) / 16;
        gemv16_wmma_f32<<<(tiles + 7) / 8, 256, 0, stream>>>(
            W_hh, hidden, b_hh, gh, 3 * H_DIM, H_DIM, 0);
    }
    // 5) GRU gate -> h_new
    gru_gate<<<(H_DIM + 255) / 256, 256, 0, stream>>>(gi, gh, hidden, out_h, hn);

    // 6) raw logits = W_out @ h_new + b_out   (50257 x 1024) -> d_out[0..V)
    {
        int tiles = (V_DIM + 15) / 16;
        gemv16_wmma_f32<<<(tiles + 7) / 8, 256, 0, stream>>>(
            W_out, hn, b_out, out, V_DIM, H_DIM, 0);
    }
    // 7) log-softmax in place
    lse_partial<<<LSE_BLOCKS, 256, 0, stream>>>(out, V_DIM, part);
    lse_final<<<1, 128, 0, stream>>>(part, LSE_BLOCKS, fin);
    lse_normalize<<<256, 256, 0, stream>>>(out, fin, V_DIM);
}